// DilateLoss_15101105012968
// MI455X (gfx1250) — compile-verified
//
#include <hip/hip_runtime.h>
#include <hip/hip_bf16.h>

// ---------------------------------------------------------------------------
// DILATE loss (soft-DTW shape + temporal terms) for B=128, N=256, D=1, f32.
// MI455X plan: WMMA f32 16x16x4 builds Delta (rank-1 cost -> one WMMA per
// 16x16 tile); wavefront DP per batch (1 workgroup / WGP), diagonals in LDS,
// R kept L2-resident in workspace; fused E*(i-j)^2 reduction in backward.
// ---------------------------------------------------------------------------

#define NN 256           // sequence length
#define RDIM 258         // (N+2) bordered R matrix
#define BIGF 1.0e9f      // reference's finite +inf border
#define NEGBIG (-1.0e30f)
#define GAMMA_C 0.01f
#define INVG 100.0f
#define ALPHA_C 0.5f

typedef float v2f __attribute__((ext_vector_type(2)));
typedef float v8f __attribute__((ext_vector_type(8)));

// ---------------------------------------------------------------------------
// Kernel 1: Delta[b][m][q] = (t[m]-o[q])^2 via V_WMMA_F32_16X16X4_F32.
// A(16x4): K0 = t[m], K1 = t[m]^2 (lanes 16..31 hold K2/K3 -> zero)
// B(4x16): K0 = -2*o[q], K1 = 1
// C      : o[q]^2 broadcast  =>  D = t^2 - 2 t o + o^2.
// One wave per (batch, 16-row strip); 16 WMMAs across the column tiles.
// ---------------------------------------------------------------------------
__global__ __launch_bounds__(32)
void dilate_delta_wmma(const float* __restrict__ targets,
                       const float* __restrict__ outputs,
                       float* __restrict__ delta)
{
    const int bid  = blockIdx.x;
    const int b    = bid >> 4;          // batch
    const int tr   = bid & 15;          // tile row (16 rows)
    const int lane = threadIdx.x;       // wave32
    const int l15  = lane & 15;

    const float* t = targets + (size_t)b * NN;
    const float* o = outputs + (size_t)b * NN;
    float* dst     = delta   + (size_t)b * NN * NN;

    const float lo16 = (lane < 16) ? 1.0f : 0.0f;   // zero K2/K3 lanes
    const int   m    = tr * 16 + l15;
    const float tv   = t[m];

    v2f a;
    a.x = tv * lo16;          // K0 = t
    a.y = tv * tv * lo16;     // K1 = t^2

    const int row0 = tr * 16 + ((lane >> 4) << 3);  // +8 for lanes 16..31

    for (int qt = 0; qt < 16; ++qt) {
        const int   q  = qt * 16 + l15;
        const float ov = o[q];

        v2f bm;
        bm.x = -2.0f * ov * lo16;   // K0 = -2*o
        bm.y = lo16;                // K1 = 1

        const float oq2 = ov * ov;
        v8f c;
#pragma unroll
        for (int r = 0; r < 8; ++r) c[r] = oq2;

        v8f d = __builtin_amdgcn_wmma_f32_16x16x4_f32(
            /*neg_a=*/false, a, /*neg_b=*/false, bm,
            /*c_mod=*/(short)0, c, /*reuse_a=*/false, /*reuse_b=*/false);

#pragma unroll
        for (int r = 0; r < 8; ++r)
            dst[(size_t)(row0 + r) * NN + q] = d[r];
    }
}

// ---------------------------------------------------------------------------
// softmin3 = -gamma*logsumexp(-x/gamma), stable form.
// ---------------------------------------------------------------------------
__device__ __forceinline__ float softmin3(float a, float b, float c)
{
    const float m = fminf(a, fminf(b, c));
    const float s = __expf((m - a) * INVG) +
                    __expf((m - b) * INVG) +
                    __expf((m - c) * INVG);
    return m - GAMMA_C * __logf(s);
}

// ---------------------------------------------------------------------------
// Kernel 2: forward soft-DTW wavefront per batch. Live diagonals in LDS;
// full R streamed to workspace (bordered with -1e30 for the backward pass).
// ---------------------------------------------------------------------------
__global__ __launch_bounds__(256)
void dilate_sdtw_fwd(const float* __restrict__ delta,
                     float* __restrict__ R,
                     float* __restrict__ vals)
{
    const int b   = blockIdx.x;
    const int tid = threadIdx.x;
    const float* D = delta + (size_t)b * NN * NN;
    float* Rb      = R     + (size_t)b * RDIM * RDIM;

    __shared__ float dbuf[3][RDIM];

    // borders of R needed by backward: row/col N+1 = -inf (finite large-neg)
    for (int i = tid; i <= NN + 1; i += blockDim.x) {
        Rb[(size_t)i * RDIM + (NN + 1)] = NEGBIG;
        Rb[(size_t)(NN + 1) * RDIM + i] = NEGBIG;
    }
    if (tid == 0) {
        dbuf[0][0] = 0.0f;              // R[0][0]
        dbuf[1][0] = BIGF;              // R[0][1]
        dbuf[1][1] = BIGF;              // R[1][0]
    }
    __syncthreads();

    int p0 = 0, p1 = 1, p2 = 2;         // diag d-2, d-1, d
    for (int d = 2; d <= 2 * NN; ++d) {
        const int ilo = (d - NN > 1) ? (d - NN) : 1;
        const int ihi = (d - 1 < NN) ? (d - 1) : NN;
        const int i   = ilo + tid;
        if (i <= ihi) {
            const int j  = d - i;
            const float diag = dbuf[p0][i - 1];
            const float up   = dbuf[p1][i - 1];
            const float left = dbuf[p1][i];
            const float r = D[(size_t)(i - 1) * NN + (j - 1)]
                          + softmin3(diag, up, left);
            dbuf[p2][i] = r;
            Rb[(size_t)i * RDIM + j] = r;
            if (j + 8 <= NN)   // stream-ahead along the row direction
                __builtin_prefetch(&D[(size_t)(i - 1) * NN + (j - 1 + 8)], 0, 0);
            if (d == 2 * NN && i == NN) {      // R[N][N]
                vals[b] = r;
                Rb[(size_t)(NN + 1) * RDIM + (NN + 1)] = r;  // R[N+1][N+1]=R[N][N]
            }
        }
        if (tid == 0 && d <= NN) {      // INF borders R[0][d], R[d][0]
            dbuf[p2][0] = BIGF;
            dbuf[p2][d] = BIGF;
        }
        __syncthreads();
        const int tp = p0; p0 = p1; p1 = p2; p2 = tp;
    }
}

// ---------------------------------------------------------------------------
// Kernel 3: backward DP (alignment matrix E) fused with sum(E * (i-j)^2).
// E never materialized: only three diagonals live in LDS.
// ---------------------------------------------------------------------------
__global__ __launch_bounds__(256)
void dilate_sdtw_bwd(const float* __restrict__ delta,
                     const float* __restrict__ R,
                     float* __restrict__ tsum)
{
    const int b   = blockIdx.x;
    const int tid = threadIdx.x;
    const float* D  = delta + (size_t)b * NN * NN;
    const float* Rb = R     + (size_t)b * RDIM * RDIM;

    __shared__ float ebuf[3][RDIM];
    __shared__ float red[256];

    if (tid == 0) {
        ebuf[0][NN + 1] = 1.0f;   // E[N+1][N+1]
        ebuf[1][NN]     = 0.0f;   // E[N][N+1]
        ebuf[1][NN + 1] = 0.0f;   // E[N+1][N]
    }
    __syncthreads();

    float acc = 0.0f;
    int p0 = 0, p1 = 1, p2 = 2;         // diag d+2, d+1, d
    for (int d = 2 * NN; d >= 2; --d) {
        const int ilo = (d - NN > 1) ? (d - NN) : 1;
        const int ihi = (d - 1 < NN) ? (d - 1) : NN;
        const int i   = ilo + tid;
        if (i <= ihi) {
            const int j = d - i;
            const float rij = Rb[(size_t)i * RDIM + j];
            const float rdn = Rb[(size_t)(i + 1) * RDIM + j];
            const float rrt = Rb[(size_t)i * RDIM + (j + 1)];
            const float rdg = Rb[(size_t)(i + 1) * RDIM + (j + 1)];
            const float ddn = (i + 1 <= NN) ? D[(size_t)i * NN + (j - 1)] : 0.0f;
            const float drt = (j + 1 <= NN) ? D[(size_t)(i - 1) * NN + j] : 0.0f;
            const float ddg = (i + 1 <= NN && j + 1 <= NN)
                              ? D[(size_t)i * NN + j] : 0.0f;
            const float wa = __expf((rdn - rij - ddn) * INVG);
            const float wb = __expf((rrt - rij - drt) * INVG);
            const float wc = __expf((rdg - rij - ddg) * INVG);
            const float e  = wa * ebuf[p1][i + 1]
                           + wb * ebuf[p1][i]
                           + wc * ebuf[p0][i + 1];
            ebuf[p2][i] = e;
            const float dij = (float)(i - j);
            acc += e * dij * dij;
        }
        if (tid == 0 && d >= NN + 2) {  // zero borders E[d-N-1][N+1], E[N+1][*]
            ebuf[p2][d - NN - 1] = 0.0f;
            ebuf[p2][NN + 1]     = 0.0f;
        }
        __syncthreads();
        const int tp = p0; p0 = p1; p1 = p2; p2 = tp;
    }

    red[tid] = acc;
    __syncthreads();
    for (int s = 128; s > 0; s >>= 1) {
        if (tid < s) red[tid] += red[tid + s];
        __syncthreads();
    }
    if (tid == 0) tsum[b] = red[0];
}

// ---------------------------------------------------------------------------
// Kernel 4: final reduction over batches -> (L_dilate, L_shape, L_temporal).
// ---------------------------------------------------------------------------
__global__ __launch_bounds__(128)
void dilate_finish(const float* __restrict__ vals,
                   const float* __restrict__ tsum,
                   float* __restrict__ out, int B)
{
    __shared__ float sv[128], st[128];
    const int tid = threadIdx.x;
    float v = 0.0f, t = 0.0f;
    for (int i = tid; i < B; i += blockDim.x) { v += vals[i]; t += tsum[i]; }
    sv[tid] = v; st[tid] = t;
    __syncthreads();
    for (int s = 64; s > 0; s >>= 1) {
        if (tid < s) { sv[tid] += sv[tid + s]; st[tid] += st[tid + s]; }
        __syncthreads();
    }
    if (tid == 0) {
        const float L_shape = sv[0] / (float)B;
        const float L_temp  = st[0] / ((float)B * (float)NN * (float)NN);
        out[0] = ALPHA_C * L_shape + (1.0f - ALPHA_C) * L_temp;
        out[1] = L_shape;
        out[2] = L_temp;
    }
}

// ---------------------------------------------------------------------------
extern "C" void kernel_launch(void* const* d_in, const int* in_sizes, int n_in,
                              void* d_out, int out_size, void* d_ws, size_t ws_size,
                              hipStream_t stream)
{
    const float* targets = (const float*)d_in[0];   // [B, N, 1]
    const float* outputs = (const float*)d_in[1];   // [B, N, 1]
    float* out = (float*)d_out;                     // 3 scalars
    const int B = in_sizes[0] / NN;                 // D == 1

    float* ws = (float*)d_ws;
    const size_t deltaElems = (size_t)B * NN * NN;
    const size_t rElems     = (size_t)B * RDIM * RDIM;
    float* dDelta = ws;
    float* dR     = ws + deltaElems;
    float* dVals  = dR + rElems;
    float* dTsum  = dVals + B;

    dilate_delta_wmma<<<dim3(B * 16), dim3(32), 0, stream>>>(targets, outputs, dDelta);
    dilate_sdtw_fwd <<<dim3(B), dim3(256), 0, stream>>>(dDelta, dR, dVals);
    dilate_sdtw_bwd <<<dim3(B), dim3(256), 0, stream>>>(dDelta, dR, dTsum);
    dilate_finish   <<<dim3(1), dim3(128), 0, stream>>>(dVals, dTsum, out, B);
}